// RangeViewQueryAndGroup_52682068853178
// MI455X (gfx1250) — compile-verified
//
#include <hip/hip_runtime.h>

// Problem constants (from reference)
#define NQ      65536     // M queries
#define NPTS    262144    // N points
#define NCH     32        // C feature channels
#define HH      64        // RV_H
#define WW      2048      // RV_W
#define PPTS    2         // P per pixel
#define NS      16        // NSAMPLE
#define NCAND   130       // (2*2+1) * (2*6+1) * 2 = 5*13*2
#define R2      4.0f      // RADIUS^2
#define OUTCH   35        // 3 + C
#define OUTSTR  560       // 35*16 floats per query

typedef __attribute__((ext_vector_type(2))) float v2f;
typedef __attribute__((ext_vector_type(8))) float v8f;

struct WaveLds {
    int   sel[NS];        // selected point indices (resolved)
    float tile[16 * 48];  // 16 samples x 35 channels, row stride 48
};

__global__ __launch_bounds__(256)
void rvq_group_kernel(const float* __restrict__ xyz,
                      const float* __restrict__ feat,
                      const float* __restrict__ qxyz,
                      const int*   __restrict__ qcoord,
                      const int*   __restrict__ rvmap,
                      float*       __restrict__ out)
{
    __shared__ WaveLds lds[8];
    const int lane = threadIdx.x & 31;
    const int wid  = threadIdx.x >> 5;
    const int q    = blockIdx.x * 8 + wid;   // one wave32 per query
    WaveLds* W = &lds[wid];

    // Query center + RV coordinate (wave-uniform values, broadcast via cache)
    const int   b  = qcoord[q * 3 + 0];
    const int   r  = qcoord[q * 3 + 1];
    const int   c  = qcoord[q * 3 + 2];
    const float qx = qxyz[q * 3 + 0];
    const float qy = qxyz[q * 3 + 1];
    const float qz = qxyz[q * 3 + 2];

    // ---- Phase 1: scan 130 candidates in order, keep first 16 valid ----
    // candidate k -> dh = k/26 - 2, dw = (k/2)%13 - 6, p = k&1  (row-major scan order)
    int nsel = 0;
    for (int t = 0; t < 5 && nsel < NS; ++t) {   // nsel is wave-uniform -> uniform exit
        const int k = t * 32 + lane;
        bool valid = false;
        int  cand  = 0;
        if (k < NCAND) {
            const int pix = k >> 1;
            const int p   = k & 1;
            const int dh  = pix / 13 - 2;
            const int dw  = pix % 13 - 6;
            const int rr  = r + dh, cc = c + dw;
            const bool okp = (rr >= 0) & (rr < HH) & (cc >= 0) & (cc < WW);
            const int rcl = min(max(rr, 0), HH - 1);
            const int ccl = min(max(cc, 0), WW - 1);
            cand = rvmap[(((b * HH + rcl) * WW) + ccl) * PPTS + p];
            if (okp && cand >= 0) {
                const float* px = xyz + (size_t)cand * 3;
                const float dx = px[0] - qx, dy = px[1] - qy, dz = px[2] - qz;
                valid = (dx * dx + dy * dy + dz * dz) <= R2;
            }
        }
        // in-order compaction: ballot + prefix popcount (wave32)
        const unsigned long long m = __ballot(valid) & 0xffffffffull;
        const int pre  = __popcll(m & ((1ull << lane) - 1ull));
        const int slot = nsel + pre;
        if (valid && slot < NS) W->sel[slot] = cand;
        nsel += __popcll(m);
    }

    asm volatile("s_wait_dscnt 0" ::: "memory");
    __builtin_amdgcn_wave_barrier();

    const bool empty = (nsel == 0);
    const int  nk    = min(nsel, NS);
    // resolve fill rule: slots >= nsel take sel[0]; empty -> index 0
    if (lane < NS) {
        int v;
        if (empty)          v = 0;
        else if (lane < nk) v = W->sel[lane];   // lane0 rewrites identical value: benign
        else                v = W->sel[0];
        W->sel[lane] = v;
    }
    asm volatile("s_wait_dscnt 0" ::: "memory");
    __builtin_amdgcn_wave_barrier();

    // ---- Phase 2: gather 16 samples x 35 channels into LDS tile ----
    for (int e = lane; e < 16 * OUTCH; e += 32) {
        const int s  = e / OUTCH;
        const int ch = e - s * OUTCH;
        const int id = W->sel[s];
        float v;
        if (ch < 3) v = xyz[(size_t)id * 3 + ch] - ((ch == 0) ? qx : (ch == 1) ? qy : qz);
        else        v = feat[(size_t)id * NCH + (ch - 3)];
        W->tile[s * 48 + ch] = v;
    }
    asm volatile("s_wait_dscnt 0" ::: "memory");
    __builtin_amdgcn_wave_barrier();

    // ---- Phase 3: transpose tile (samples x channels -> channels x samples) ----
#if __has_builtin(__builtin_amdgcn_wmma_f32_16x16x4_f32)
    // WMMA-identity transpose: D[m][n] = sum_k A[m][k]*B[k][n] with
    // A[m][k] = tile[sample=koff+k][chan=chb*16+m], B[k][n] = (n == k+koff)
    //  => D[chan][sample]. A layout: lane m (0-15 both halves), half picks K{0,1}/{2,3}.
    //  B layout: lane n (0-15 both halves), half picks K{0,1}/{2,3}. D: vgpr v -> row v+8*half, col lane&15.
    const int half = lane >> 4;
    const int ln   = lane & 15;
    for (int chb = 0; chb < 3; ++chb) {
        v8f d = {};
        #pragma unroll
        for (int koff = 0; koff < 16; koff += 4) {
            const int s0 = koff + 2 * half;
            v2f a, bf;
            a.x  = W->tile[s0 * 48 + chb * 16 + ln];
            a.y  = W->tile[(s0 + 1) * 48 + chb * 16 + ln];
            bf.x = (ln == 2 * half + 0 + koff) ? 1.0f : 0.0f;
            bf.y = (ln == 2 * half + 1 + koff) ? 1.0f : 0.0f;
            d = __builtin_amdgcn_wmma_f32_16x16x4_f32(
                    false, a, false, bf, (short)0, d, false, false);
        }
        const int rows = (chb == 2) ? 3 : 16;   // block 2 holds channels 32..34 only
        #pragma unroll
        for (int v = 0; v < 8; ++v) {
            const int mrow = v + 8 * half;
            if (mrow < rows) {
                const float val = empty ? 0.0f : d[v];
                out[(size_t)q * OUTSTR + (chb * 16 + mrow) * 16 + ln] = val;
            }
        }
    }
#else
    // Fallback: conflict-free LDS transpose (stride 48 -> distinct banks)
    for (int e = lane; e < OUTSTR; e += 32) {
        const int ch = e >> 4, s = e & 15;
        out[(size_t)q * OUTSTR + e] = empty ? 0.0f : W->tile[s * 48 + ch];
    }
#endif
}

extern "C" void kernel_launch(void* const* d_in, const int* in_sizes, int n_in,
                              void* d_out, int out_size, void* d_ws, size_t ws_size,
                              hipStream_t stream) {
    (void)in_sizes; (void)n_in; (void)out_size; (void)d_ws; (void)ws_size;
    const float* xyz    = (const float*)d_in[0];
    const float* feat   = (const float*)d_in[1];
    const float* qxyz   = (const float*)d_in[2];
    const int*   qcoord = (const int*)d_in[3];
    const int*   rvmap  = (const int*)d_in[4];
    float* out = (float*)d_out;

    dim3 grid(NQ / 8);   // 8 waves (queries) per 256-thread block
    dim3 block(256);
    hipLaunchKernelGGL(rvq_group_kernel, grid, block, 0, stream,
                       xyz, feat, qxyz, qcoord, rvmap, out);
}